// Pointnet_SA_58317065945844
// MI455X (gfx1250) — compile-verified
//
#include <hip/hip_runtime.h>

// Problem constants (must match reference)
#define B_  16
#define N_  4096
#define C_  64
#define S_  1024
#define K_  32
#define D0  67     // 3 + C  (input dim of layer 0)
#define D0P 68     // padded to multiple of 4
#define F0  64
#define F1  64
#define F2  128
#define LDA 68     // LDS row stride for activation buffer A

typedef __attribute__((ext_vector_type(2))) float v2f;
typedef __attribute__((ext_vector_type(8))) float v8f;

// Only check for the builtin in the DEVICE pass -- amdgcn builtins are never
// visible to the host compilation.
#if defined(__HIP_DEVICE_COMPILE__) && !__has_builtin(__builtin_amdgcn_wmma_f32_16x16x4_f32)
#error "device pass: wmma_f32_16x16x4_f32 builtin not available"
#endif

// ---------------------------------------------------------------------------
// Kernel 1: farthest point sampling, one block per batch.
// Each thread keeps its 16 points + running min-dist in registers.
// Writes new_xyz (B,S,3) directly into d_out.
// ---------------------------------------------------------------------------
__global__ __launch_bounds__(256) void fps_kernel(const float* __restrict__ xyz,
                                                  float* __restrict__ out_xyz) {
    __shared__ float rd[256];
    __shared__ int   ri[256];
    const int b = blockIdx.x, tid = threadIdx.x;

    float px[16], py[16], pz[16], dst[16];
#pragma unroll
    for (int j = 0; j < 16; ++j) {
        const int i = tid + j * 256;
        const float* p = xyz + ((size_t)b * N_ + i) * 3;
        px[j] = p[0]; py[j] = p[1]; pz[j] = p[2];
        dst[j] = 1e10f;
    }

    int last = 0;
    for (int t = 0; t < S_; ++t) {
        // broadcast coords of 'last' via (uniform) scalar global load
        const float* lp = xyz + ((size_t)b * N_ + last) * 3;
        const float lx = lp[0], ly = lp[1], lz = lp[2];
        if (tid == 0) {
            float* o = out_xyz + ((size_t)b * S_ + t) * 3;
            o[0] = lx; o[1] = ly; o[2] = lz;
        }
        float lm = -1.0f; int li = 0;
#pragma unroll
        for (int j = 0; j < 16; ++j) {
            const float dx = px[j] - lx, dy = py[j] - ly, dz = pz[j] - lz;
            const float d = dx * dx + dy * dy + dz * dz;
            const float nd = fminf(dst[j], d);
            dst[j] = nd;
            if (nd > lm) { lm = nd; li = tid + j * 256; }
        }
        rd[tid] = lm; ri[tid] = li;
        __syncthreads();
        for (int s = 128; s > 0; s >>= 1) {
            if (tid < s) {
                const float od = rd[tid + s]; const int oi = ri[tid + s];
                if (od > rd[tid] || (od == rd[tid] && oi < ri[tid])) {
                    rd[tid] = od; ri[tid] = oi;
                }
            }
            __syncthreads();
        }
        last = ri[0];
        __syncthreads();
    }
}

// ---------------------------------------------------------------------------
// Kernel 2: exact 32-NN per centroid. One wave (32 lanes) per query,
// 2 waves per block. Distances cached in LDS, then 32 rounds of
// wave-wide lexicographic min + extraction.
// ---------------------------------------------------------------------------
__global__ __launch_bounds__(64) void knn_kernel(const float* __restrict__ xyz,
                                                 const float* __restrict__ new_xyz,
                                                 int* __restrict__ knn) {
    __shared__ float sd[2 * N_];   // 32 KB
    const int tid  = threadIdx.x;
    const int lane = tid & 31;
    const int wid  = tid >> 5;
    const int q    = blockIdx.x * 2 + wid;       // global query id
    const int b    = q >> 10;                    // q / S_
    float* base = sd + wid * N_;

    const float* nq = new_xyz + (size_t)q * 3;   // uniform per wave
    const float qx = nq[0], qy = nq[1], qz = nq[2];

    const float* pb = xyz + (size_t)b * N_ * 3;
#pragma unroll 4
    for (int j = 0; j < N_ / 32; ++j) {
        const int p = j * 32 + lane;
        const float dx = pb[p * 3 + 0] - qx;
        const float dy = pb[p * 3 + 1] - qy;
        const float dz = pb[p * 3 + 2] - qz;
        base[p] = dx * dx + dy * dy + dz * dz;
    }

    for (int r = 0; r < K_; ++r) {
        float bd = 3.4e38f; int bi = 0x7fffffff;
        for (int j = 0; j < N_ / 32; ++j) {
            const int p = j * 32 + lane;
            const float d = base[p];
            if (d < bd || (d == bd && p < bi)) { bd = d; bi = p; }
        }
#pragma unroll
        for (int off = 16; off > 0; off >>= 1) {
            const float od = __shfl_xor(bd, off, 32);
            const int   oi = __shfl_xor(bi, off, 32);
            if (od < bd || (od == bd && oi < bi)) { bd = od; bi = oi; }
        }
        if ((bi & 31) == lane) base[bi] = 3.4e38f;  // extract winner
        if (lane == 0) knn[(size_t)q * K_ + r] = bi;
    }
}

// ---------------------------------------------------------------------------
// Kernel 3: gather + 3-layer MLP (fp32 WMMA 16x16x4) + max-pool over K.
// Block = 128 threads (4 waves) = 64 rows = 2 centroids.
// Activations ping-pong through LDS; each wave only touches its own 16 rows.
// ---------------------------------------------------------------------------
__global__ __launch_bounds__(128) void mlp_kernel(
    const float* __restrict__ xyz, const float* __restrict__ pts,
    const float* __restrict__ w0, const float* __restrict__ bs0,
    const float* __restrict__ w1, const float* __restrict__ bs1,
    const float* __restrict__ w2, const float* __restrict__ bs2,
    const float* __restrict__ new_xyz, const int* __restrict__ knn,
    float* __restrict__ out_pts) {

    __shared__ float bufA[64 * LDA];   // 17.4 KB
    __shared__ float bufB[64 * F0];    // 16 KB
    __shared__ int   sidx[64];
    __shared__ float cc[2][3];
    __shared__ float part[4 * F2];     // 2 KB

    const int tid  = threadIdx.x;
    const int lane = tid & 31;
    const int wid  = tid >> 5;
    const int b    = blockIdx.x >> 9;       // S_/2 = 512 groups per batch
    const int cg   = blockIdx.x & 511;
    const int s0   = cg * 2;

    __builtin_prefetch(w0 + tid, 0, 0);
    __builtin_prefetch(w2 + tid, 0, 0);

    if (tid < 64) sidx[tid] = knn[((size_t)b * S_ + s0 + (tid >> 5)) * K_ + (tid & 31)];
    if (tid < 6) {
        const int c = tid / 3, d = tid % 3;
        cc[c][d] = new_xyz[((size_t)b * S_ + s0 + c) * 3 + d];
    }
    __syncthreads();

    // Build X tile: 64 rows x 68 cols  ([dx,dy,dz, feat0..63, 0])
    for (int e = tid; e < 64 * D0P; e += 128) {
        const int r = e / D0P, c = e - r * D0P;
        const int ci = r >> 5;
        const int gi = sidx[r];
        float v;
        if (c < 3) {
            v = xyz[((size_t)b * N_ + gi) * 3 + c] - cc[ci][c];
        } else if (c < D0) {
            v = pts[((size_t)b * N_ + gi) * C_ + (c - 3)];
        } else {
            v = 0.0f;
        }
        bufA[r * LDA + c] = v;
    }
    __syncthreads();

#if defined(__HIP_DEVICE_COMPILE__)
    const int R     = wid * 16;
    const int rowA  = R + (lane & 15);
    const int khalf = (lane >> 4) * 2;   // lanes 0-15 -> K0/K1, lanes 16-31 -> K2/K3
    const int rw    = R + (lane >> 4) * 8;
    const int nsub  = lane & 15;

    // ---- Layer 0: (64 x 67) @ (67 x 64) -> bufB, ReLU ----
    for (int nt = 0; nt < F0 / 16; ++nt) {
        const int col = nt * 16 + nsub;
        v8f acc = {};
        for (int k0 = 0; k0 < D0P; k0 += 4) {
            const int kk = k0 + khalf;
            v2f a; a.x = bufA[rowA * LDA + kk]; a.y = bufA[rowA * LDA + kk + 1];
            v2f w; w.x = (kk     < D0) ? w0[kk       * F0 + col] : 0.0f;
                   w.y = (kk + 1 < D0) ? w0[(kk + 1) * F0 + col] : 0.0f;
            acc = __builtin_amdgcn_wmma_f32_16x16x4_f32(false, a, false, w,
                                                        (short)0, acc, false, false);
        }
        const float bias = bs0[col];
#pragma unroll
        for (int j = 0; j < 8; ++j)
            bufB[(rw + j) * F0 + col] = fmaxf(acc[j] + bias, 0.0f);
    }
    __syncthreads();

    // ---- Layer 1: (64 x 64) @ (64 x 64) -> bufA, ReLU ----
    for (int nt = 0; nt < F1 / 16; ++nt) {
        const int col = nt * 16 + nsub;
        v8f acc = {};
        for (int k0 = 0; k0 < F0; k0 += 4) {
            const int kk = k0 + khalf;
            v2f a; a.x = bufB[rowA * F0 + kk]; a.y = bufB[rowA * F0 + kk + 1];
            v2f w; w.x = w1[kk * F1 + col]; w.y = w1[(kk + 1) * F1 + col];
            acc = __builtin_amdgcn_wmma_f32_16x16x4_f32(false, a, false, w,
                                                        (short)0, acc, false, false);
        }
        const float bias = bs1[col];
#pragma unroll
        for (int j = 0; j < 8; ++j)
            bufA[(rw + j) * LDA + col] = fmaxf(acc[j] + bias, 0.0f);
    }
    __syncthreads();

    // ---- Layer 2: (64 x 64) @ (64 x 128), ReLU, max over rows fused ----
    for (int nt = 0; nt < F2 / 16; ++nt) {
        const int col = nt * 16 + nsub;
        v8f acc = {};
        for (int k0 = 0; k0 < F1; k0 += 4) {
            const int kk = k0 + khalf;
            v2f a; a.x = bufA[rowA * LDA + kk]; a.y = bufA[rowA * LDA + kk + 1];
            v2f w; w.x = w2[kk * F2 + col]; w.y = w2[(kk + 1) * F2 + col];
            acc = __builtin_amdgcn_wmma_f32_16x16x4_f32(false, a, false, w,
                                                        (short)0, acc, false, false);
        }
        const float bias = bs2[col];
        float m = 0.0f;  // post-ReLU values are >= 0
#pragma unroll
        for (int j = 0; j < 8; ++j)
            m = fmaxf(m, fmaxf(acc[j] + bias, 0.0f));
        m = fmaxf(m, __shfl_xor(m, 16, 32));      // combine row halves
        if (lane < 16) part[wid * F2 + col] = m;  // max over this wave's 16 rows
    }
    __syncthreads();
#endif  // __HIP_DEVICE_COMPILE__

    // Combine the two waves of each centroid, write (B,S,128) output
    for (int e = tid; e < 2 * F2; e += 128) {
        const int c = e >> 7, col = e & 127;
        const float v = fmaxf(part[(2 * c) * F2 + col], part[(2 * c + 1) * F2 + col]);
        out_pts[((size_t)b * S_ + s0 + c) * F2 + col] = v;
    }
}

// ---------------------------------------------------------------------------
extern "C" void kernel_launch(void* const* d_in, const int* in_sizes, int n_in,
                              void* d_out, int out_size, void* d_ws, size_t ws_size,
                              hipStream_t stream) {
    const float* xyz    = (const float*)d_in[0];
    const float* points = (const float*)d_in[1];
    const float* w0     = (const float*)d_in[2];
    const float* b0     = (const float*)d_in[3];
    const float* w1     = (const float*)d_in[4];
    const float* b1     = (const float*)d_in[5];
    const float* w2     = (const float*)d_in[6];
    const float* b2     = (const float*)d_in[7];

    float* out_xyz = (float*)d_out;                          // (B,S,3)
    float* out_pts = (float*)d_out + (size_t)B_ * S_ * 3;    // (B,S,128)
    int*   knn     = (int*)d_ws;                             // (B,S,K) = 2 MB

    fps_kernel<<<B_, 256, 0, stream>>>(xyz, out_xyz);
    knn_kernel<<<(B_ * S_) / 2, 64, 0, stream>>>(xyz, out_xyz, knn);
    mlp_kernel<<<(B_ * S_) / 2, 128, 0, stream>>>(xyz, points, w0, b0, w1, b1, w2, b2,
                                                  out_xyz, knn, out_pts);
}